// CTCLayer_47536698032824
// MI455X (gfx1250) — compile-verified
//
#include <hip/hip_runtime.h>
#include <cstdint>

// ---------------------------------------------------------------------------
// CTC batch cost (Keras ctc_batch_cost semantics) for MI455X / gfx1250.
//
// Shapes: B=128, T=160, C=6000, U=32, S=2U+1=65, blank=C-1, eps=1e-7, NEG=-1e30
//
// Roofline: the DP only touches blank + <=32 label columns of y_pred, i.e.
// B*T*33 = 675840 scattered 4B reads (<=~86 MB of 128B line fills ~ 4us at
// 23.3 TB/s). Compute (128*160*65 log-add-exps) is negligible. So the kernel
// is a hardware gather (CDNA5 async data mover: per-lane global addr ->
// per-lane LDS addr, ASYNCcnt-tracked, non-temporal) followed by a tiny
// in-LDS alpha recursion.
// ---------------------------------------------------------------------------

namespace {
constexpr int BATCH = 128;
constexpr int T     = 160;
constexpr int C     = 6000;
constexpr int U     = 32;
constexpr int S     = 2 * U + 1;   // 65 extended states
constexpr int NCOL  = U + 1;       // 33 distinct symbol columns: [blank, l0..l31]
constexpr int BLANK = C - 1;
constexpr float NEGF = -1e30f;
constexpr float EPSF = 1e-7f;
constexpr int BLK   = 256;         // 8 wave32s
}

__device__ __forceinline__ float logaddexpf_(float a, float b) {
    // matches jnp.logaddexp incl. the (NEG,NEG) -> NEG + log(2) corner
    float mx = fmaxf(a, b);
    float mn = fminf(a, b);
    return mx + log1pf(expf(mn - mx));
}

__global__ __launch_bounds__(BLK) void ctc_loss_kernel(
    const int*   __restrict__ y_true,        // [B,U]
    const float* __restrict__ y_pred,        // [B,T,C] probabilities
    const int*   __restrict__ input_length,  // [B,1]
    const int*   __restrict__ label_length,  // [B,1]
    float*       __restrict__ out)           // [B,1]
{
    __shared__ float          g[T * NCOL];   // gathered raw probabilities (21120 B)
    __shared__ int            labels[U];
    __shared__ unsigned char  skip[S];
    __shared__ float          alpha[S + 2];  // [0..1] are permanent NEG padding

    const int b   = blockIdx.x;
    const int tid = threadIdx.x;

    if (tid < U)     labels[tid] = y_true[b * U + tid];
    if (tid < S + 2) alpha[tid]  = NEGF;
    __syncthreads();

    // skip[s]=1 for odd s>=3 iff labels[k]!=labels[k-1], k=(s-1)/2  (ref: 3::2)
    if (tid < S) {
        bool sk = false;
        if ((tid & 1) && tid >= 3) {
            int k = tid >> 1;                 // == (tid-1)/2 for odd tid
            sk = (labels[k] != labels[k - 1]);
        }
        skip[tid] = (unsigned char)sk;
    }

    // ---- Phase 1: async non-temporal gather  g[t][u] = y_pred[b,t,cls(u)] --
    // cls(0)=blank, cls(u>=1)=labels[u-1].  Per-lane scattered global reads
    // deposited straight into LDS by the async data mover (ASYNCcnt), marked
    // NT: each fetched line is used ~once, don't let it displace cache.
    {
        const float* base = y_pred + (size_t)b * (size_t)T * (size_t)C;
        for (int idx = tid; idx < T * NCOL; idx += BLK) {
            int t = idx / NCOL;
            int u = idx - t * NCOL;
            int cls = (u == 0) ? BLANK : labels[u - 1];
            const float* src = base + (size_t)t * (size_t)C + (size_t)cls;
            // low 32 bits of a flat LDS pointer == wave-relative LDS byte offset
            unsigned dst = (unsigned)(uintptr_t)(&g[idx]);
            asm volatile("global_load_async_to_lds_b32 %0, %1, off th:TH_LOAD_NT"
                         :: "v"(dst), "v"(src)
                         : "memory");
        }
        // wait for this wave's async transfers, then block-wide barrier makes
        // every wave's slice of g[] visible.
        asm volatile("s_wait_asynccnt 0x0" ::: "memory");
    }
    __syncthreads();

    const int in_len = input_length[b];

    // t = 0 initialization: only s=0 (blank) and s=1 (first label)
    if (tid == 0) {
        alpha[2 + 0] = logf(g[0] + EPSF);     // g[0*NCOL+0] : blank column
        alpha[2 + 1] = logf(g[1] + EPSF);     // g[0*NCOL+1] : first label
    }
    __syncthreads();

    // ---- Phase 2: alpha recursion ------------------------------------------
    for (int t = 1; t < T; ++t) {
        float v = 0.0f;
        if (tid < S) {
            float a  = alpha[2 + tid];        // alpha[s]
            float a1 = alpha[1 + tid];        // alpha[s-1] (pad -> NEG)
            float a2 = alpha[0 + tid];        // alpha[s-2] (pad -> NEG)
            float m  = logaddexpf_(a, a1);
            if (skip[tid]) m = logaddexpf_(m, a2);
            int u = (tid & 1) ? (1 + (tid >> 1)) : 0;   // state -> column
            float lp = logf(g[t * NCOL + u] + EPSF);
            float nv = m + lp;
            v = (t < in_len) ? nv : a;        // freeze past input_length
        }
        __syncthreads();
        if (tid < S) alpha[2 + tid] = v;
        __syncthreads();
    }

    // loss = -logaddexp(alpha[2L], alpha[2L-1])
    if (tid == 0) {
        int end = 2 * label_length[b];
        out[b] = -logaddexpf_(alpha[2 + end], alpha[1 + end]);
    }
}

extern "C" void kernel_launch(void* const* d_in, const int* /*in_sizes*/, int /*n_in*/,
                              void* d_out, int /*out_size*/, void* /*d_ws*/, size_t /*ws_size*/,
                              hipStream_t stream) {
    const int*   y_true       = (const int*)d_in[0];
    const float* y_pred       = (const float*)d_in[1];
    const int*   input_length = (const int*)d_in[2];
    const int*   label_length = (const int*)d_in[3];
    hipLaunchKernelGGL(ctc_loss_kernel, dim3(BATCH), dim3(BLK), 0, stream,
                       y_true, y_pred, input_length, label_length, (float*)d_out);
}